// SSS_Block_38465727103444
// MI455X (gfx1250) — compile-verified
//
#include <hip/hip_runtime.h>
#include <hip/hip_bf16.h>
#include <math.h>

// Problem constants (match reference)
#define DIMC   192
#define DI     384          // d_inner
#define LSEQ   4096         // H*W
#define BATCH  2
#define DTR    12           // dt_rank
#define NST    16           // d_state
#define KX     44           // DTR + 2*NST
#define NXZ    768          // 2*DI
#define BL_TOT 8192         // BATCH*LSEQ

typedef float v2f __attribute__((ext_vector_type(2)));
typedef float v8f __attribute__((ext_vector_type(8)));

__device__ __forceinline__ float softplus_f(float x) {
    // stable logaddexp(x, 0) — matches jax.nn.softplus
    return fmaxf(x, 0.f) + log1pf(expf(-fabsf(x)));
}
__device__ __forceinline__ float silu_f(float x) {
    return x / (1.f + expf(-x));
}

// ---------------------------------------------------------------------------
// 0) Transpose W_out (192,384) -> W_outT (384,192) so GEMM-out B loads coalesce
// ---------------------------------------------------------------------------
__global__ void k_transpose_wout(const float* __restrict__ Wout,
                                 float* __restrict__ WoutT) {
    int i = blockIdx.x * blockDim.x + threadIdx.x;
    if (i >= DI * DIMC) return;
    int k = i / DIMC, c = i % DIMC;
    WoutT[i] = Wout[c * DI + k];
}

// ---------------------------------------------------------------------------
// 1) Input projection: XZ[bl, 0:768] = xf[bl, 0:192] @ W_in  (fp32 WMMA)
//    x is (B, C, L): A is column-major in (bl, k) -> contiguous lane loads.
//    Each wave: one 16(M) x 64(N) strip, K-loop of v_wmma_f32_16x16x4_f32.
// ---------------------------------------------------------------------------
__global__ void __launch_bounds__(128)
k_gemm_in(const float* __restrict__ X, const float* __restrict__ Win,
          float* __restrict__ XZ) {
    const int lane = threadIdx.x & 31;
    const int wv   = threadIdx.x >> 5;
    const int hi   = lane >> 4;     // selects K pair {0,1} vs {2,3}
    const int lo   = lane & 15;     // M (A) / N (B,C) index
    const int m0 = (blockIdx.x * 4 + wv) * 16;
    const int n0 = blockIdx.y * 64;
    const int b  = m0 / LSEQ;
    const int l0 = m0 % LSEQ;

    const float* Ap = X + (size_t)b * DIMC * LSEQ + l0 + lo;   // + k*LSEQ
    v8f acc0 = {}, acc1 = {}, acc2 = {}, acc3 = {};

    for (int k0 = 0; k0 < DIMC; k0 += 4) {
        const int ka = k0 + 2 * hi;
        v2f a;
        a.x = Ap[(size_t)ka * LSEQ];
        a.y = Ap[(size_t)(ka + 1) * LSEQ];
        const float* Bp = Win + (size_t)ka * NXZ + n0 + lo;
        v2f b0, b1, b2, b3;
        b0.x = Bp[0];   b0.y = Bp[NXZ + 0];
        b1.x = Bp[16];  b1.y = Bp[NXZ + 16];
        b2.x = Bp[32];  b2.y = Bp[NXZ + 32];
        b3.x = Bp[48];  b3.y = Bp[NXZ + 48];
        acc0 = __builtin_amdgcn_wmma_f32_16x16x4_f32(false, a, false, b0, (short)0, acc0, false, false);
        acc1 = __builtin_amdgcn_wmma_f32_16x16x4_f32(false, a, false, b1, (short)0, acc1, false, false);
        acc2 = __builtin_amdgcn_wmma_f32_16x16x4_f32(false, a, false, b2, (short)0, acc2, false, false);
        acc3 = __builtin_amdgcn_wmma_f32_16x16x4_f32(false, a, false, b3, (short)0, acc3, false, false);
    }
    #pragma unroll
    for (int r = 0; r < 8; ++r) {
        const int row = m0 + r + 8 * hi;           // C layout: M = vgpr + 8*half
        float* o = XZ + (size_t)row * NXZ + n0 + lo;
        o[0]  = acc0[r];
        o[16] = acc1[r];
        o[32] = acc2[r];
        o[48] = acc3[r];
    }
}

// ---------------------------------------------------------------------------
// 2) Depthwise causal conv (k=4, left pad 3) + SiLU: XC[bl, d]
// ---------------------------------------------------------------------------
__global__ void k_conv(const float* __restrict__ XZ, const float* __restrict__ cw,
                       const float* __restrict__ cb, float* __restrict__ XC) {
    int idx = blockIdx.x * blockDim.x + threadIdx.x;
    if (idx >= BL_TOT * DI) return;
    const int d  = idx % DI;
    const int bl = idx / DI;
    const int l  = bl % LSEQ;
    float s = cb[d];
    #pragma unroll
    for (int k = 0; k < 4; ++k) {
        const int ll = l - 3 + k;
        if (ll >= 0) s += XZ[(size_t)(bl - 3 + k) * NXZ + d] * cw[d * 4 + k];
    }
    XC[(size_t)bl * DI + d] = silu_f(s);
}

// ---------------------------------------------------------------------------
// 3) x_dbl[bl, 0:44] = XC[bl, :] @ W_x^T   (K=384, N=44 -> VALU + LDS row)
// ---------------------------------------------------------------------------
__global__ void __launch_bounds__(64)
k_xdbl(const float* __restrict__ XC, const float* __restrict__ Wx,
       float* __restrict__ XDBL) {
    __shared__ float row[DI];
    const int bl = blockIdx.x;
    for (int i = threadIdx.x; i < DI; i += 64) row[i] = XC[(size_t)bl * DI + i];
    __syncthreads();
    const int kk = threadIdx.x;
    if (kk < KX) {
        float s = 0.f;
        for (int dd = 0; dd < DI; ++dd) s += row[dd] * Wx[kk * DI + dd];
        XDBL[(size_t)bl * KX + kk] = s;
    }
}

// ---------------------------------------------------------------------------
// 4) delta[bl, d] = softplus(dt @ W_dt^T + 2*b_dt)   (K=12 -> VALU)
//    (b_dt is added twice, faithfully reproducing the reference)
// ---------------------------------------------------------------------------
__global__ void k_delta(const float* __restrict__ XDBL, const float* __restrict__ Wdt,
                        const float* __restrict__ bdt, float* __restrict__ DLT) {
    int idx = blockIdx.x * blockDim.x + threadIdx.x;
    if (idx >= BL_TOT * DI) return;
    const int d  = idx % DI;
    const int bl = idx / DI;
    float s = 2.f * bdt[d];
    #pragma unroll
    for (int r = 0; r < DTR; ++r) s += XDBL[(size_t)bl * KX + r] * Wdt[d * DTR + r];
    DLT[idx] = softplus_f(s);
}

// ---------------------------------------------------------------------------
// 5) Selective scan: 768 recurrences, one per 16-lane half-wave (lane = state n).
//    y written TRANSPOSED: YT[d][bl] -> contiguous per stream, and gives the
//    output GEMM a column-major A operand (contiguous WMMA lane loads).
// ---------------------------------------------------------------------------
__global__ void __launch_bounds__(256)
k_scan(const float* __restrict__ DLT, const float* __restrict__ XC,
       const float* __restrict__ XDBL, const float* __restrict__ XZ,
       const float* __restrict__ Alog, const float* __restrict__ Dp,
       float* __restrict__ YT) {
    const int s = blockIdx.x * 16 + (threadIdx.x >> 4);   // 0..767
    const int n = threadIdx.x & 15;                       // state index
    const int b = s / DI;
    const int d = s % DI;
    const float An = -expf(Alog[d * NST + n]);
    const float Dd = Dp[d];
    float h = 0.f;
    const size_t blBase = (size_t)b * LSEQ;
    for (int l = 0; l < LSEQ; ++l) {
        const size_t bl = blBase + l;
        const float dlt = DLT[bl * DI + d];            // broadcast in half-wave
        const float u   = XC[bl * DI + d];             // broadcast
        const float Bv  = XDBL[bl * KX + DTR + n];     // 16-lane contiguous
        const float Cv  = XDBL[bl * KX + DTR + NST + n];
        h = expf(dlt * An) * h + dlt * Bv * u;
        float p = h * Cv;
        p += __shfl_xor(p, 1, 16);                     // reduce over 16 states
        p += __shfl_xor(p, 2, 16);
        p += __shfl_xor(p, 4, 16);
        p += __shfl_xor(p, 8, 16);
        if (n == 0) {
            const float zg = XZ[bl * NXZ + DI + d];
            YT[(size_t)d * BL_TOT + bl] = (p + u * Dd) * silu_f(zg);
        }
    }
}

// ---------------------------------------------------------------------------
// 6) Output projection: out[(b*192+c)*L + l] = YT^T @ W_outT   (fp32 WMMA)
// ---------------------------------------------------------------------------
__global__ void __launch_bounds__(128)
k_gemm_out(const float* __restrict__ YT, const float* __restrict__ WoT,
           float* __restrict__ OUT) {
    const int lane = threadIdx.x & 31;
    const int wv   = threadIdx.x >> 5;
    const int hi   = lane >> 4;
    const int lo   = lane & 15;
    const int m0 = (blockIdx.x * 4 + wv) * 16;
    const int n0 = blockIdx.y * 64;
    const int b  = m0 / LSEQ;

    const float* Ap = YT + m0 + lo;                    // + k*BL_TOT
    v8f acc0 = {}, acc1 = {}, acc2 = {}, acc3 = {};

    for (int k0 = 0; k0 < DI; k0 += 4) {
        const int ka = k0 + 2 * hi;
        v2f a;
        a.x = Ap[(size_t)ka * BL_TOT];
        a.y = Ap[(size_t)(ka + 1) * BL_TOT];
        const float* Bp = WoT + (size_t)ka * DIMC + n0 + lo;
        v2f b0, b1, b2, b3;
        b0.x = Bp[0];   b0.y = Bp[DIMC + 0];
        b1.x = Bp[16];  b1.y = Bp[DIMC + 16];
        b2.x = Bp[32];  b2.y = Bp[DIMC + 32];
        b3.x = Bp[48];  b3.y = Bp[DIMC + 48];
        acc0 = __builtin_amdgcn_wmma_f32_16x16x4_f32(false, a, false, b0, (short)0, acc0, false, false);
        acc1 = __builtin_amdgcn_wmma_f32_16x16x4_f32(false, a, false, b1, (short)0, acc1, false, false);
        acc2 = __builtin_amdgcn_wmma_f32_16x16x4_f32(false, a, false, b2, (short)0, acc2, false, false);
        acc3 = __builtin_amdgcn_wmma_f32_16x16x4_f32(false, a, false, b3, (short)0, acc3, false, false);
    }
    #pragma unroll
    for (int r = 0; r < 8; ++r) {
        const int m = m0 + r + 8 * hi;
        const int l = m % LSEQ;
        float* o = OUT + ((size_t)(b * DIMC + n0 + lo)) * LSEQ + l;
        o[(size_t)0 * LSEQ]  = acc0[r];
        o[(size_t)16 * LSEQ] = acc1[r];
        o[(size_t)32 * LSEQ] = acc2[r];
        o[(size_t)48 * LSEQ] = acc3[r];
    }
}

// ---------------------------------------------------------------------------
extern "C" void kernel_launch(void* const* d_in, const int* in_sizes, int n_in,
                              void* d_out, int out_size, void* d_ws, size_t ws_size,
                              hipStream_t stream) {
    const float* x    = (const float*)d_in[0];   // (2,192,4096)
    const float* Win  = (const float*)d_in[1];   // (192,768)
    const float* cw   = (const float*)d_in[2];   // (384,4)
    const float* cb   = (const float*)d_in[3];   // (384)
    const float* Wx   = (const float*)d_in[4];   // (44,384)
    const float* Wdt  = (const float*)d_in[5];   // (384,12)
    const float* bdt  = (const float*)d_in[6];   // (384)
    const float* Alog = (const float*)d_in[7];   // (384,16)
    const float* Dp   = (const float*)d_in[8];   // (384)
    const float* Wout = (const float*)d_in[9];   // (192,384)
    float* out = (float*)d_out;

    // Workspace layout (~64.4 MB of fp32)
    float* ws   = (float*)d_ws;
    float* XZ   = ws;                               // 8192*768
    float* XC   = XZ   + (size_t)BL_TOT * NXZ;      // 8192*384
    float* XDBL = XC   + (size_t)BL_TOT * DI;       // 8192*44
    float* DLT  = XDBL + (size_t)BL_TOT * KX;       // 8192*384
    float* YT   = DLT  + (size_t)BL_TOT * DI;       // 384*8192 (transposed)
    float* WoT  = YT   + (size_t)DI * BL_TOT;       // 384*192

    k_transpose_wout<<<(DI * DIMC + 255) / 256, 256, 0, stream>>>(Wout, WoT);
    k_gemm_in<<<dim3(BL_TOT / 64, NXZ / 64), 128, 0, stream>>>(x, Win, XZ);
    k_conv<<<(BL_TOT * DI + 255) / 256, 256, 0, stream>>>(XZ, cw, cb, XC);
    k_xdbl<<<BL_TOT, 64, 0, stream>>>(XC, Wx, XDBL);
    k_delta<<<(BL_TOT * DI + 255) / 256, 256, 0, stream>>>(XDBL, Wdt, bdt, DLT);
    k_scan<<<(BATCH * DI) / 16, 256, 0, stream>>>(DLT, XC, XDBL, XZ, Alog, Dp, YT);
    k_gemm_out<<<dim3(BL_TOT / 64, DIMC / 64), 128, 0, stream>>>(YT, WoT, out);
}